// CollageRenderer_30528627540025
// MI455X (gfx1250) — compile-verified
//
#include <hip/hip_runtime.h>
#include <stdint.h>

#define CS 1024
#define TT 512
#define NS 64
#define PLANE (TT * TT)       // one texture channel plane (elements)
#define OPLANE (CS * CS)      // one output channel plane (elements)

// ---------------------------------------------------------------------------
// Pre-pass: interleave planar [4][4][TT][TT] textures into [4][TT][TT] float4
// (RGBA) so each bilinear corner in the main kernel is a single b128 load.
// ---------------------------------------------------------------------------
__global__ __launch_bounds__(256) void interleave_kernel(
    const float* __restrict__ tex,    // [4][4][TT][TT]
    float4*      __restrict__ rgba)   // [4][TT][TT]
{
    const int i = blockIdx.x * 256 + threadIdx.x;   // 0 .. 4*PLANE-1
    const int t = i >> 18;                          // texture id (PLANE = 2^18)
    const int p = i & (PLANE - 1);                  // texel index
    const float* b = tex + (size_t)t * 4 * PLANE + p;
    float4 v;
    v.x = b[0];
    v.y = b[PLANE];
    v.z = b[2 * PLANE];
    v.w = b[3 * PLANE];
    rgba[i] = v;
}

// ---------------------------------------------------------------------------
// Main pass: one thread per pixel, fold all 64 shape composites in registers,
// write RGB once. Shape params staged into LDS via CDNA5 async global->LDS
// loads (ASYNCcnt / s_wait_asynccnt). RGBA=true samples the interleaved
// scratch buffer (1 x b128 per corner); RGBA=false samples planar textures.
// ---------------------------------------------------------------------------
template <bool RGBA>
__global__ __launch_bounds__(256) void collage_kernel(
    const int*    __restrict__ shape_type,   // [NS]
    const float*  __restrict__ params,       // [NS][9]
    const float*  __restrict__ tex,          // [4][4][TT][TT] (planar path)
    const float4* __restrict__ rgba,         // [4][TT][TT]    (rgba path)
    float*        __restrict__ out)          // [3][CS][CS]
{
    __shared__ float raw[NS * 9];   // raw params staged via async copy
    __shared__ int   rst[NS];       // shape_type staged via async copy
    __shared__ float sc[NS * 9];    // derived: A,B,C,D, tx,ty, r,g,b
    __shared__ int   sbase[NS];     // texture base offset (elements / float4s)

    const int tid = threadIdx.y * 16 + threadIdx.x;   // 0..255

    // Warm L2 with the texture stacks (global_prefetch_b8).
    if (tid < 4) {
        if (RGBA) __builtin_prefetch(rgba + tid * PLANE, 0, 3);
        else      __builtin_prefetch(tex + tid * 4 * PLANE, 0, 3);
    }

    // ---- CDNA5 async global -> LDS staging (tracked by ASYNCcnt) ----
    for (int i = tid; i < NS * 9; i += 256) {
        unsigned lo = (unsigned)(uint64_t)(&raw[i]);       // LDS byte offset
        uint64_t ga = (uint64_t)(params + i);
        asm volatile("global_load_async_to_lds_b32 %0, %1, off"
                     :: "v"(lo), "v"(ga) : "memory");
    }
    if (tid < NS) {
        unsigned lo = (unsigned)(uint64_t)(&rst[tid]);
        uint64_t ga = (uint64_t)(shape_type + tid);
        asm volatile("global_load_async_to_lds_b32 %0, %1, off"
                     :: "v"(lo), "v"(ga) : "memory");
    }
    asm volatile("s_wait_asynccnt 0" ::: "memory");
    __syncthreads();

    // ---- derive per-shape affine constants (64 threads, one per shape) ----
    if (tid < NS) {
        const float* p = &raw[tid * 9];
        float txc = p[0] * (float)CS;
        float tyc = p[1] * (float)CS;
        float W   = (float)CS * (0.05f + 0.95f * p[2]);
        float H   = (float)CS * (0.05f + 0.95f * p[3]);
        float asn = p[4], acs = p[5];
        float nrm = sqrtf(asn * asn + acs * acs) + 1e-8f;
        float c = acs / nrm, s = asn / nrm;
        float iW = (float)TT / W, iH = (float)TT / H;
        float* q = &sc[tid * 9];
        q[0] =  c * iW;   // tex_x = A*dx + B*dy + off
        q[1] =  s * iW;
        q[2] = -s * iH;   // tex_y = C*dx + D*dy + off
        q[3] =  c * iH;
        q[4] = txc;
        q[5] = tyc;
        q[6] = p[6];      // r
        q[7] = p[7];      // g
        q[8] = p[8];      // b
        sbase[tid] = rst[tid] * (RGBA ? PLANE : 4 * PLANE);
    }
    __syncthreads();

    // ---- per-pixel composite over all shapes, in registers ----
    const int x = blockIdx.x * 16 + threadIdx.x;
    const int y = blockIdx.y * 16 + threadIdx.y;
    const float px = (float)x + 0.5f;
    const float py = (float)y + 0.5f;
    const float toff = 0.5f * (float)TT - 0.5f;

    float cr = 1.0f, cg = 1.0f, cb = 1.0f;   // canvas starts at 1.0

    #pragma unroll 1
    for (int k = 0; k < NS; ++k) {
        const float* q = &sc[k * 9];
        float dx = px - q[4];
        float dy = py - q[5];
        float txf = fmaf(q[0], dx, fmaf(q[1], dy, toff));
        float tyf = fmaf(q[2], dx, fmaf(q[3], dy, toff));

        // whole 2x2 bilinear footprint off-texture -> alpha 0 -> no-op
        if (txf < -1.0f || txf >= (float)TT || tyf < -1.0f || tyf >= (float)TT)
            continue;

        float x0f = floorf(txf), y0f = floorf(tyf);
        float wx = txf - x0f,    wy = tyf - y0f;
        int x0 = (int)x0f, y0 = (int)y0f;
        int x1 = x0 + 1,   y1 = y0 + 1;

        // fold per-corner validity into the weights (invalid sample == 0)
        bool vx0 = (x0 >= 0) & (x0 < TT);
        bool vx1 = (x1 >= 0) & (x1 < TT);
        bool vy0 = (y0 >= 0) & (y0 < TT);
        bool vy1 = (y1 >= 0) & (y1 < TT);
        float w00 = (vx0 & vy0) ? (1.0f - wx) * (1.0f - wy) : 0.0f;
        float w01 = (vx1 & vy0) ? wx * (1.0f - wy)          : 0.0f;
        float w10 = (vx0 & vy1) ? (1.0f - wx) * wy          : 0.0f;
        float w11 = (vx1 & vy1) ? wx * wy                   : 0.0f;

        int xc0 = min(max(x0, 0), TT - 1), xc1 = min(max(x1, 0), TT - 1);
        int yc0 = min(max(y0, 0), TT - 1), yc1 = min(max(y1, 0), TT - 1);
        int o00 = yc0 * TT + xc0, o01 = yc0 * TT + xc1;
        int o10 = yc1 * TT + xc0, o11 = yc1 * TT + xc1;

        float a, sr, sg, sb;
        if (RGBA) {
            // one global_load_b128 per corner
            const float4* tb = rgba + sbase[k];
            float4 v00 = tb[o00];
            float4 v01 = tb[o01];
            float4 v10 = tb[o10];
            float4 v11 = tb[o11];
            sr = w00 * v00.x + w01 * v01.x + w10 * v10.x + w11 * v11.x;
            sg = w00 * v00.y + w01 * v01.y + w10 * v10.y + w11 * v11.y;
            sb = w00 * v00.z + w01 * v01.z + w10 * v10.z + w11 * v11.z;
            a  = w00 * v00.w + w01 * v01.w + w10 * v10.w + w11 * v11.w;
        } else {
            const float* tr = tex + sbase[k];       // R plane
            const float* tg = tr + PLANE;           // G plane
            const float* tb = tr + 2 * PLANE;       // B plane
            const float* ta = tr + 3 * PLANE;       // A plane
            a  = w00 * ta[o00] + w01 * ta[o01] + w10 * ta[o10] + w11 * ta[o11];
            sr = w00 * tr[o00] + w01 * tr[o01] + w10 * tr[o10] + w11 * tr[o11];
            sg = w00 * tg[o00] + w01 * tg[o01] + w10 * tg[o10] + w11 * tg[o11];
            sb = w00 * tb[o00] + w01 * tb[o01] + w10 * tb[o10] + w11 * tb[o11];
        }

        float ia = 1.0f - a;
        cr = cr * ia + sr * q[6] * a;
        cg = cg * ia + sg * q[7] * a;
        cb = cb * ia + sb * q[8] * a;
    }

    const int pix = y * CS + x;
    out[pix]              = cr;
    out[OPLANE + pix]     = cg;
    out[2 * OPLANE + pix] = cb;
}

extern "C" void kernel_launch(void* const* d_in, const int* in_sizes, int n_in,
                              void* d_out, int out_size, void* d_ws, size_t ws_size,
                              hipStream_t stream) {
    const int*   st     = (const int*)d_in[0];     // shape_type [64]
    const float* params = (const float*)d_in[1];   // params [64,9]
    const float* tex    = (const float*)d_in[2];   // textures [4,4,512,512]
    float*       out    = (float*)d_out;           // [3,1024,1024]

    dim3 block(16, 16);
    dim3 grid(CS / 16, CS / 16);

    const size_t need = (size_t)4 * PLANE * sizeof(float4);   // 16 MB
    if (ws_size >= need && d_ws != nullptr) {
        float4* rgba = (float4*)d_ws;
        interleave_kernel<<<(4 * PLANE) / 256, 256, 0, stream>>>(
            (const float*)d_in[2], rgba);
        collage_kernel<true><<<grid, block, 0, stream>>>(
            st, params, tex, rgba, out);
    } else {
        collage_kernel<false><<<grid, block, 0, stream>>>(
            st, params, tex, (const float4*)nullptr, out);
    }
}